// PaniniGenomicGrammar_52321291600386
// MI455X (gfx1250) — compile-verified
//
#include <hip/hip_runtime.h>

#define BATCH    2
#define SEQ      2048
#define EMBD     1024
#define RANK     8
#define VOCAB    32000
#define NBUCKETS 16
#define BSIZE    (SEQ / NBUCKETS)   // 128
#define NTILES   (VOCAB / 16)       // 2000
#define MTILES   ((BATCH * SEQ) / 16) // 256

typedef __attribute__((ext_vector_type(2))) float v2f;
typedef __attribute__((ext_vector_type(4))) float v4f;
typedef __attribute__((ext_vector_type(8))) float v8f;

// ---------------------------------------------------------------------------
// Phase 1: low[token][r] = sum_e hidden[token][e] * A[bucket(t)][e][r]
// One wave (32 lanes) per token. 4096 tokens total.
// ---------------------------------------------------------------------------
__global__ __launch_bounds__(256) void panini_lowrank_kernel(
    const float* __restrict__ hidden, const float* __restrict__ A,
    float* __restrict__ low) {
  const int lane  = threadIdx.x & 31;
  const int wave  = threadIdx.x >> 5;
  const int token = blockIdx.x * 8 + wave;      // 0 .. BATCH*SEQ-1
  const int t     = token & (SEQ - 1);
  const int bucket = min(t / BSIZE, NBUCKETS - 1);

  const float* hrow = hidden + (size_t)token * EMBD;
  const float* Ab   = A + (size_t)bucket * EMBD * RANK;

  float acc[RANK];
#pragma unroll
  for (int r = 0; r < RANK; ++r) acc[r] = 0.0f;

  for (int e = lane; e < EMBD; e += 32) {
    const float h = hrow[e];
    const v4f a0 = *(const v4f*)(Ab + (size_t)e * RANK);
    const v4f a1 = *(const v4f*)(Ab + (size_t)e * RANK + 4);
#pragma unroll
    for (int r = 0; r < 4; ++r) acc[r]     += h * a0[r];
#pragma unroll
    for (int r = 0; r < 4; ++r) acc[4 + r] += h * a1[r];
  }

  // wave32 tree reduction: after this every lane holds the full sums
#pragma unroll
  for (int r = 0; r < RANK; ++r) {
#pragma unroll
    for (int m = 16; m >= 1; m >>= 1)
      acc[r] += __shfl_xor(acc[r], m, 32);
  }

  if (lane == 0) {
    v4f o0 = {acc[0], acc[1], acc[2], acc[3]};
    v4f o1 = {acc[4], acc[5], acc[6], acc[7]};
    *(v4f*)(low + (size_t)token * RANK)     = o0;
    *(v4f*)(low + (size_t)token * RANK + 4) = o1;
  }
}

// ---------------------------------------------------------------------------
// Phase 2: out[m, v] = alpha * low[m, 0:8] @ B[0:8, v]
// 16x16 output tiles via two chained V_WMMA_F32_16X16X4_F32 (K = 8 = 2 * 4).
// One wave per M-tile, striding over N-tiles.
//
// 32-bit A-operand layout (16x4, MxK): lanes 0-15 -> M=lane, VGPR0=K0,VGPR1=K1;
//                                      lanes 16-31 -> M=lane-16, K2/K3.
// 32-bit B-operand layout (4x16, KxN): lanes 0-15 -> N=lane, rows K0/K1;
//                                      lanes 16-31 -> rows K2/K3.
// C/D layout: VGPR i -> (M=i, N=lane) for lanes 0-15, (M=8+i, N=lane-16) else.
// ---------------------------------------------------------------------------
__global__ __launch_bounds__(256) void panini_vocab_wmma_kernel(
    const float* __restrict__ low, const float* __restrict__ Bm,
    const float* __restrict__ alpha_p, float* __restrict__ out) {
  const int lane   = threadIdx.x & 31;
  const int wave   = threadIdx.x >> 5;
  const int m_tile = blockIdx.y;                // 0 .. MTILES-1
  const float alpha = *alpha_p;

  const int m  = lane & 15;
  const int kb = (lane >> 4) << 1;              // 0 (lanes 0-15) or 2 (lanes 16-31)

  // A operand: the 16x8 low tile, split into K=[0..3] and K=[4..7]
  const float* lrow = low + (size_t)(m_tile * 16 + m) * RANK;
  v2f a_lo, a_hi;
  a_lo.x = lrow[kb + 0];
  a_lo.y = lrow[kb + 1];
  a_hi.x = lrow[kb + 4];
  a_hi.y = lrow[kb + 5];

  const int rbase = (lane >> 4) * 8;            // output row group per half-wave

  for (int n_tile = blockIdx.x * 8 + wave; n_tile < NTILES;
       n_tile += gridDim.x * 8) {
    const int col = n_tile * 16 + (lane & 15);

    v2f b_lo, b_hi;
    b_lo.x = Bm[(size_t)(kb + 0) * VOCAB + col];
    b_lo.y = Bm[(size_t)(kb + 1) * VOCAB + col];
    b_hi.x = Bm[(size_t)(kb + 4) * VOCAB + col];
    b_hi.y = Bm[(size_t)(kb + 5) * VOCAB + col];

    v8f c = {};
    c = __builtin_amdgcn_wmma_f32_16x16x4_f32(
        /*neg_a=*/false, a_lo, /*neg_b=*/false, b_lo,
        /*c_mod=*/(short)0, c, /*reuse_a=*/false, /*reuse_b=*/false);
    c = __builtin_amdgcn_wmma_f32_16x16x4_f32(
        false, a_hi, false, b_hi, (short)0, c, false, false);

    float* ocol = out + (size_t)(m_tile * 16 + rbase) * VOCAB + col;
#pragma unroll
    for (int i = 0; i < 8; ++i)
      ocol[(size_t)i * VOCAB] = alpha * c[i];
  }
}

extern "C" void kernel_launch(void* const* d_in, const int* in_sizes, int n_in,
                              void* d_out, int out_size, void* d_ws, size_t ws_size,
                              hipStream_t stream) {
  const float* hidden = (const float*)d_in[0];   // [2, 2048, 1024]
  const float* A      = (const float*)d_in[1];   // [16, 1024, 8]
  const float* Bm     = (const float*)d_in[2];   // [8, 32000]
  const float* alpha  = (const float*)d_in[3];   // scalar
  float* out = (float*)d_out;                    // [2, 2048, 32000]
  float* low = (float*)d_ws;                     // [4096, 8] scratch (128 KB)

  // Phase 1: 4096 tokens, 8 waves per 256-thread block -> 512 blocks
  panini_lowrank_kernel<<<dim3((BATCH * SEQ) / 8), 256, 0, stream>>>(hidden, A, low);

  // Phase 2: grid (50 N-groups x 256 M-tiles), 8 waves/block.
  // 50*8 = 400 wave slots covering 2000 N-tiles in exactly 5 uniform iterations.
  panini_vocab_wmma_kernel<<<dim3(50, MTILES), 256, 0, stream>>>(low, Bm, alpha, out);
}